// AggregationAndDiscriminationLoss_87806311400034
// MI455X (gfx1250) — compile-verified
//
#include <hip/hip_runtime.h>

#define NPIX (896*896)
#define NCH 4
#define NIMG 16
#define ACC_STRIDE 80   // per image: A[16], Bk[16], Cc[16], cK[16], cT[16]
#define NBIN 17         // bins indexed by raw label 0..16; bin 0 discarded

typedef __attribute__((ext_vector_type(2))) float v2f;
typedef __attribute__((ext_vector_type(8))) float v8f;

__device__ __forceinline__ void lds_fadd(float* p, float v) {
    __hip_atomic_fetch_add(p, v, __ATOMIC_RELAXED, __HIP_MEMORY_SCOPE_WORKGROUP);
}
__device__ __forceinline__ void glb_fadd(float* p, float v) {
    __hip_atomic_fetch_add(p, v, __ATOMIC_RELAXED, __HIP_MEMORY_SCOPE_AGENT);
}

__global__ void zero_ws_kernel(float* acc, int* kmax) {
    int i = blockIdx.x * blockDim.x + threadIdx.x;
    if (i < NIMG * ACC_STRIDE) acc[i] = 0.f;
    if (i < NIMG) kmax[i] = 0;
}

// Streaming segment-sum phase: bandwidth-bound (~308 MB total -> ~13us @ 23.3 TB/s).
// Branch-free scatter: 17 bins indexed by raw label; label-0 traffic lands in a
// discarded slot, and the T==K overlap test is a value-select, not a branch.
__global__ __launch_bounds__(256) void accum_kernel(
    const float* __restrict__ sim, const int* __restrict__ Tl,
    const int* __restrict__ Kl, float* __restrict__ acc, int* __restrict__ kmax)
{
    __shared__ float bins[8][5][NBIN];   // [wave][quantity][bin]
    __shared__ int wmax;
    const int tid = threadIdx.x;
    const int b = blockIdx.y;

    for (int i = tid; i < 8 * 5 * NBIN; i += 256) (&bins[0][0][0])[i] = 0.f;
    if (tid == 0) wmax = 0;
    __syncthreads();

    const long long nb = (long long)b * NPIX;
    const int g = (blockIdx.x * 256 + tid) * 4;   // 4 pixels per thread, exact tiling
    const float* p = sim + (long long)b * NCH * NPIX + g;
    float4 s0 = *(const float4*)(p);
    float4 s1 = *(const float4*)(p + NPIX);
    float4 s2 = *(const float4*)(p + 2 * NPIX);
    float4 s3 = *(const float4*)(p + 3 * NPIX);
    int4 t4 = *(const int4*)(Tl + nb + g);
    int4 k4 = *(const int4*)(Kl + nb + g);

    float (*mb)[NBIN] = bins[tid >> 5];   // this wave's private copy

    int mk = max(max(k4.x, k4.y), max(k4.z, k4.w));
    atomicMax(&wmax, mk);

#define DOPIX(X) {                                                              \
    float q = fmaf(s0.X, s0.X, fmaf(s1.X, s1.X, fmaf(s2.X, s2.X, s3.X*s3.X)));  \
    int t = t4.X, k = k4.X;                                                     \
    lds_fadd(&mb[0][t], q);                                                     \
    lds_fadd(&mb[4][t], 1.f);                                                   \
    lds_fadd(&mb[1][k], q);                                                     \
    lds_fadd(&mb[3][k], 1.f);                                                   \
    lds_fadd(&mb[2][k], (t == k) ? q : 0.f); }
    DOPIX(x) DOPIX(y) DOPIX(z) DOPIX(w)
#undef DOPIX

    __syncthreads();
    if (tid < 80) {   // tid = quantity*16 + (bin-1), matches acc layout
        float s = 0.f;
        #pragma unroll
        for (int w = 0; w < 8; ++w) s += bins[w][tid >> 4][(tid & 15) + 1];
        glb_fadd(&acc[b * ACC_STRIDE + tid], s);
    }
    if (tid == 128) atomicMax(&kmax[b], wmax);
}

// One wave32, EXEC all ones. Pairwise matrix S = D*1^T + 1*D^T computed with a
// single v_wmma_f32_16x16x4_f32 per image (A = [D, 1, 0, 0], B = [1; D^T; 0; 0]).
__global__ __launch_bounds__(32) void finalize_kernel(
    const float* __restrict__ acc, const int* __restrict__ kmax, float* __restrict__ out)
{
    const int lane = threadIdx.x;
    const int m = lane & 15;      // column / label index - 1
    const int hi = lane >> 4;
    float Lagg = 0.f, Ldis = 0.f;

    for (int b = 0; b < NIMG; ++b) {
        const float* a = acc + b * ACC_STRIDE;
        const int nk = kmax[b];
        const float A  = a[m];
        const float Bk = a[16 + m];
        const float Cc = a[32 + m];
        const float cK = a[48 + m];
        const float cT = a[64 + m];

        const float dk = cK + 0.001f;
        const float Dm = Bk / (dk * dk);   // D[m], identical for row and col use

        // A 16x4 layout: lanes 0-15 hold K=0 (vgpr0) and K=1 (vgpr1); lanes 16-31 hold K=2,3 (zero).
        v2f av; av.x = hi ? 0.f : Dm;  av.y = hi ? 0.f : 1.f;
        // B 4x16 layout: vgpr0 = row K=0 (lanes 0-15) / K=2 (lanes 16-31); vgpr1 = K=1 / K=3.
        v2f bv; bv.x = hi ? 0.f : 1.f; bv.y = hi ? 0.f : Dm;
        v8f c = {0.f, 0.f, 0.f, 0.f, 0.f, 0.f, 0.f, 0.f};
        v8f d = __builtin_amdgcn_wmma_f32_16x16x4_f32(
            false, av, false, bv, (short)0, c, false, false);

        // Discrimination: element (row i = v + 8*hi, col j = m) holds S = D_i + D_j.
        float ds = 0.f;
        #pragma unroll
        for (int v = 0; v < 8; ++v) {
            int i = v + 8 * hi;
            int j = m;
            if (i < j && (j + 1) <= nk) {
                float S = d[v];
                float pn = S > 0.f ? sqrtf(S) : 0.f;
                float dn = 3.0f - pn;
                ds += logf(fmaf(dn, dn, 1.f));
            }
        }
        for (int off = 16; off > 0; off >>= 1) ds += __shfl_xor(ds, off, 32);
        float nkf = (float)nk;
        if (nk > 1) Ldis += ds / (nkf * (nkf - 1.f));

        // Aggregation: 16 terms, lanes 0-15.
        float at = 0.f;
        if (hi == 0 && (m + 1) <= nk) {
            float den = cK + 1.f;
            float x = A + Bk / (den * den) - 2.f * Cc / den;
            float sq = x > 0.f ? sqrtf(x) : 0.f;
            float nrm = sq - 0.5f;
            at = logf(fmaf(nrm, nrm, 1.f)) / (cT + 1.f);
        }
        for (int off = 16; off > 0; off >>= 1) at += __shfl_xor(at, off, 32);
        Lagg += at;
    }

    if (lane == 0) { out[0] = Lagg; out[1] = Ldis; }
}

extern "C" void kernel_launch(void* const* d_in, const int* in_sizes, int n_in,
                              void* d_out, int out_size, void* d_ws, size_t ws_size,
                              hipStream_t stream) {
    const float* sim = (const float*)d_in[0];
    const int*   Tl  = (const int*)d_in[1];
    const int*   Kl  = (const int*)d_in[2];
    float* out = (float*)d_out;
    float* acc = (float*)d_ws;
    int* kmax = (int*)((char*)d_ws + (size_t)NIMG * ACC_STRIDE * sizeof(float));

    zero_ws_kernel<<<(NIMG * ACC_STRIDE + 255) / 256, 256, 0, stream>>>(acc, kmax);

    dim3 grid(NPIX / (4 * 256), NIMG);   // 784 x 16 blocks, exact cover
    accum_kernel<<<grid, 256, 0, stream>>>(sim, Tl, Kl, acc, kmax);

    finalize_kernel<<<1, 32, 0, stream>>>(acc, kmax, out);
}